// cfsmn_cell_54451595379268
// MI455X (gfx1250) — compile-verified
//
#include <hip/hip_runtime.h>

typedef __attribute__((ext_vector_type(2))) float v2f;
typedef __attribute__((ext_vector_type(8))) float v8f;

#define T_DIM 2048
#define N_DIM 16
#define I_DIM 1024
#define H_DIM 2048
#define O_DIM 1024
#define K_TAPS 12
#define STRIDE_C 2
#define LMEM 10
#define M_DIM (T_DIM * N_DIM) /* 32768 */

#define BM 128
#define BN 128
#define BK 32
#define LDSK (BK + 4) /* 36 floats: float4-aligned, bank-spread */
#define THREADS 256

// ---------------------------------------------------------------------------
// CDNA5 async global->LDS fill (tracked by ASYNCcnt), 16B per lane.
// %0 = LDS byte address (low 32 bits of generic pointer), %1 = global addr.
// ---------------------------------------------------------------------------
__device__ __forceinline__ void async_ld16(unsigned lds_addr,
                                           const float* gaddr) {
  asm volatile("global_load_async_to_lds_b128 %0, %1, off"
               :
               : "v"(lds_addr), "v"(gaddr)
               : "memory");
}

// ---------------------------------------------------------------------------
// Async double-buffered f32 WMMA GEMM: C[M,N] = A[M,K] * B[N,K]^T
//   block tile 128x128x32, 8 waves, each wave 32x64 via 2x4 v8f accumulators.
//   SCATTER_NT: row r = t*16+n, write C[n][t][col] (GEMM1 -> P layout [N,T,H])
//   BIAS_RELU : C = relu(C + bias[col])                   (GEMM2 epilogue)
// ---------------------------------------------------------------------------
template <int KDIM, bool SCATTER_NT, bool BIAS_RELU>
__launch_bounds__(THREADS)
__global__ void wmma_gemm_kernel(const float* __restrict__ A,
                                 const float* __restrict__ Bm,
                                 const float* __restrict__ bias,
                                 float* __restrict__ C) {
  constexpr int LDC = SCATTER_NT ? H_DIM : O_DIM;
  constexpr int NSTAGE = KDIM / BK;

  __shared__ __align__(16) float As[2][BM * LDSK];
  __shared__ __align__(16) float Bs[2][BN * LDSK];

  const int tid  = threadIdx.x;
  const int m0   = blockIdx.x * BM;
  const int n0   = blockIdx.y * BN;
  const int lrow = tid >> 3;          // 0..31
  const int lcol = (tid & 7) * 4;     // 0,4,...,28
  const int wid  = tid >> 5;
  const int lane = tid & 31;
  const int wm   = wid & 3;           // 32-row slice
  const int wn   = wid >> 2;          // 64-col slice
  const int lr   = lane & 15;
  const int lh   = lane >> 4;

  const float* Ablk = A + (size_t)m0 * KDIM;
  const float* Bblk = Bm + (size_t)n0 * KDIM;

  unsigned ldsA[2], ldsB[2];
  ldsA[0] = (unsigned)(uintptr_t)&As[0][0];
  ldsA[1] = (unsigned)(uintptr_t)&As[1][0];
  ldsB[0] = (unsigned)(uintptr_t)&Bs[0][0];
  ldsB[1] = (unsigned)(uintptr_t)&Bs[1][0];

  v8f acc[2][4] = {};

  // Prologue: fill stage 0 into buffer 0.
#pragma unroll
  for (int r = 0; r < 4; ++r) {
    const int row = lrow + r * 32;
    const unsigned loff = (unsigned)(row * LDSK + lcol) * 4u;
    async_ld16(ldsA[0] + loff, Ablk + (size_t)row * KDIM + lcol);
    async_ld16(ldsB[0] + loff, Bblk + (size_t)row * KDIM + lcol);
  }

  for (int s = 0; s < NSTAGE; ++s) {
    const int cur = s & 1;
    if (s + 1 < NSTAGE) {
      // Issue next stage into the other buffer, then wait only for stage s:
      // async ops complete in order, so ASYNCcnt<=8 => first 8 are done.
      const int kb = (s + 1) * BK;
#pragma unroll
      for (int r = 0; r < 4; ++r) {
        const int row = lrow + r * 32;
        const unsigned loff = (unsigned)(row * LDSK + lcol) * 4u;
        async_ld16(ldsA[cur ^ 1] + loff, Ablk + (size_t)row * KDIM + kb + lcol);
        async_ld16(ldsB[cur ^ 1] + loff, Bblk + (size_t)row * KDIM + kb + lcol);
      }
      asm volatile("s_wait_asynccnt 0x8" ::: "memory");
    } else {
      asm volatile("s_wait_asynccnt 0x0" ::: "memory");
    }
    __syncthreads();

    const float* Asb = &As[cur][0];
    const float* Bsb = &Bs[cur][0];
#pragma unroll
    for (int kk = 0; kk < BK; kk += 4) {
      v2f a[2], b[4];
#pragma unroll
      for (int i = 0; i < 2; ++i)
        a[i] = *reinterpret_cast<const v2f*>(
            &Asb[(wm * 32 + i * 16 + lr) * LDSK + kk + lh * 2]);
#pragma unroll
      for (int j = 0; j < 4; ++j)
        b[j] = *reinterpret_cast<const v2f*>(
            &Bsb[(wn * 64 + j * 16 + lr) * LDSK + kk + lh * 2]);
#pragma unroll
      for (int i = 0; i < 2; ++i)
#pragma unroll
        for (int j = 0; j < 4; ++j)
          acc[i][j] = __builtin_amdgcn_wmma_f32_16x16x4_f32(
              false, a[i], false, b[j], (short)0, acc[i][j], false, false);
    }
    __syncthreads();  // everyone done reading buf cur before it is refilled
  }

  // Epilogue (D layout: M = 8*lh + v per VGPR v, N = lane%16).
#pragma unroll
  for (int i = 0; i < 2; ++i)
#pragma unroll
    for (int j = 0; j < 4; ++j) {
      const int col = n0 + wn * 64 + j * 16 + lr;
      float bb = 0.f;
      if constexpr (BIAS_RELU) bb = bias[col];
#pragma unroll
      for (int v = 0; v < 8; ++v) {
        const int r = m0 + wm * 32 + i * 16 + lh * 8 + v;
        float val = acc[i][j][v];
        if constexpr (BIAS_RELU) {
          val += bb;
          val = val > 0.f ? val : 0.f;
        }
        size_t idx;
        if constexpr (SCATTER_NT) {
          const int t = r >> 4;
          const int n = r & 15;
          idx = ((size_t)n * T_DIM + t) * LDC + col;
        } else {
          idx = (size_t)r * LDC + col;
        }
        C[idx] = val;
      }
    }
}

// ---------------------------------------------------------------------------
// Conv + BN partial stats. p_hatt[n,t,h] = p + sum_k w[k,h]*p[n,t+(k-10)*2,h]
// block: 64 h-lanes x 4 t-groups; grid (H/64, N). Deterministic partials.
// ---------------------------------------------------------------------------
__launch_bounds__(256)
__global__ void conv_stats_kernel(const float* __restrict__ P,
                                  const float* __restrict__ MW,
                                  float* __restrict__ Phat,
                                  float* __restrict__ part_sum,
                                  float* __restrict__ part_sq) {
  const int tid = threadIdx.x;
  const int h   = blockIdx.x * 64 + (tid & 63);
  const int tg  = tid >> 6;
  const int n   = blockIdx.y;

  float wk[K_TAPS];
#pragma unroll
  for (int k = 0; k < K_TAPS; ++k) wk[k] = MW[k * H_DIM + h];

  const float* pb = P + (size_t)n * T_DIM * H_DIM + h;
  float* ob = Phat + (size_t)n * T_DIM * H_DIM + h;

  float lsum = 0.f, lsq = 0.f;
  for (int t = tg; t < T_DIM; t += 4) {
    float acc = pb[(size_t)t * H_DIM];
#pragma unroll
    for (int k = 0; k < K_TAPS; ++k) {
      const int tt = t + (k - LMEM) * STRIDE_C;
      if (tt >= 0 && tt < T_DIM)
        acc = fmaf(wk[k], pb[(size_t)tt * H_DIM], acc);
    }
    ob[(size_t)t * H_DIM] = acc;
    lsum += acc;
    lsq = fmaf(acc, acc, lsq);
  }

  __shared__ float s0[256];
  __shared__ float s1[256];
  s0[tid] = lsum;
  s1[tid] = lsq;
  __syncthreads();
  if (tg == 0) {
    const float a  = s0[tid] + s0[tid + 64] + s0[tid + 128] + s0[tid + 192];
    const float b2 = s1[tid] + s1[tid + 64] + s1[tid + 128] + s1[tid + 192];
    part_sum[(size_t)n * H_DIM + h] = a;
    part_sq[(size_t)n * H_DIM + h] = b2;
  }
}

// ---------------------------------------------------------------------------
// BN fold: scale = gamma*rsqrt(var+eps), shift = beta - mean*scale.
// ---------------------------------------------------------------------------
__global__ void finalize_stats_kernel(const float* __restrict__ part_sum,
                                      const float* __restrict__ part_sq,
                                      const float* __restrict__ gamma,
                                      const float* __restrict__ beta,
                                      float* __restrict__ scale,
                                      float* __restrict__ shift) {
  const int h = blockIdx.x * blockDim.x + threadIdx.x;
  if (h >= H_DIM) return;
  float s = 0.f, q = 0.f;
  for (int n = 0; n < N_DIM; ++n) {
    s += part_sum[(size_t)n * H_DIM + h];
    q += part_sq[(size_t)n * H_DIM + h];
  }
  const float inv = 1.0f / (float)((size_t)N_DIM * T_DIM);
  const float mean = s * inv;
  const float var  = fmaf(-mean, mean, q * inv);
  const float sc   = gamma[h] * rsqrtf(var + 1e-5f);
  scale[h] = sc;
  shift[h] = fmaf(-mean, sc, beta[h]);
}

// ---------------------------------------------------------------------------
// Fold BN scale into the up-projection weights: wu_s[o,h] = wu[o,h]*scale[h].
// ---------------------------------------------------------------------------
__global__ void scale_wu_kernel(const float* __restrict__ wu,
                                const float* __restrict__ scale,
                                float* __restrict__ wu_s) {
  const size_t i = (size_t)blockIdx.x * blockDim.x + threadIdx.x;
  const int h = (int)(i & (H_DIM - 1));
  wu_s[i] = wu[i] * scale[h];
}

// ---------------------------------------------------------------------------
// Fold BN shift + bias into per-output constant: c0[o]=bu[o]+sum_h sh[h]*wu[o,h]
// ---------------------------------------------------------------------------
__global__ void bias_fold_kernel(const float* __restrict__ wu,
                                 const float* __restrict__ shift,
                                 const float* __restrict__ bu,
                                 float* __restrict__ c0) {
  const int o = blockIdx.x * blockDim.x + threadIdx.x;
  if (o >= O_DIM) return;
  float s = bu[o];
  const float* row = wu + (size_t)o * H_DIM;
  for (int h = 0; h < H_DIM; ++h) s = fmaf(shift[h], row[h], s);
  c0[o] = s;
}

// ---------------------------------------------------------------------------
extern "C" void kernel_launch(void* const* d_in, const int* in_sizes, int n_in,
                              void* d_out, int out_size, void* d_ws,
                              size_t ws_size, hipStream_t stream) {
  (void)in_sizes; (void)n_in; (void)out_size; (void)ws_size;
  const float* inputs = (const float*)d_in[0];
  const float* wv     = (const float*)d_in[1];
  const float* gamma  = (const float*)d_in[2];
  const float* beta   = (const float*)d_in[3];
  const float* wu     = (const float*)d_in[4];
  const float* bu     = (const float*)d_in[5];
  const float* mw     = (const float*)d_in[6];
  // d_in[7..9]: stride=2, l_memory_size=10, r_memory_size=1 (fixed by setup)

  float* out_h    = (float*)d_out;                          // [N,T,O]
  float* out_phat = out_h + (size_t)N_DIM * T_DIM * O_DIM;  // [N,T,H]

  float* p_ws     = (float*)d_ws;                           // [N,T,H]
  float* part_sum = p_ws + (size_t)N_DIM * T_DIM * H_DIM;   // [N,H]
  float* part_sq  = part_sum + (size_t)N_DIM * H_DIM;       // [N,H]
  float* scale    = part_sq + (size_t)N_DIM * H_DIM;        // [H]
  float* shift    = scale + H_DIM;                          // [H]
  float* wu_s     = shift + H_DIM;                          // [O,H]
  float* c0       = wu_s + (size_t)O_DIM * H_DIM;           // [O]

  // 1) p = inputs @ wv^T  (scatter to [N,T,H])
  wmma_gemm_kernel<I_DIM, true, false>
      <<<dim3(M_DIM / BM, H_DIM / BN), dim3(THREADS), 0, stream>>>(
          inputs, wv, nullptr, p_ws);
  // 2) depthwise dilated conv + residual + BN partial stats; p_hatt -> d_out
  conv_stats_kernel<<<dim3(H_DIM / 64, N_DIM), dim3(256), 0, stream>>>(
      p_ws, mw, out_phat, part_sum, part_sq);
  // 3) BN affine fold
  finalize_stats_kernel<<<dim3((H_DIM + 255) / 256), dim3(256), 0, stream>>>(
      part_sum, part_sq, gamma, beta, scale, shift);
  // 4) fold BN into GEMM2 operands
  scale_wu_kernel<<<dim3(((size_t)O_DIM * H_DIM) / 256), dim3(256), 0,
                    stream>>>(wu, scale, wu_s);
  bias_fold_kernel<<<dim3((O_DIM + 255) / 256), dim3(256), 0, stream>>>(
      wu, shift, bu, c0);
  // 5) h = relu(p_hatt @ wu_s^T + c0)
  wmma_gemm_kernel<H_DIM, false, true>
      <<<dim3(M_DIM / BM, O_DIM / BN), dim3(THREADS), 0, stream>>>(
          out_phat, wu_s, c0, out_h);
}